// PNA_23201413333400
// MI455X (gfx1250) — compile-verified
//
#include <hip/hip_runtime.h>

// PNA GNN for MI455X (gfx1250), fp32 WMMA 16x16x4 for edge/node GEMMs.
#define NN 20000      // nodes
#define NE 320000     // edges
#define FD 75         // feature dim
#define FP 80         // padded feature stride (5 x 16)
#define GG 64         // graphs
#define CCL 10        // classes
#define LL 5          // PNA layers
#define KE 152        // padded 2F (150 -> 152, mult of 4)
#define KESTR 156     // LDS row stride for A tiles (bank-conflict-free)
#define KN 976        // padded 13F (975 -> 976, mult of 4)
#define EPSV 1e-5f

typedef float v2f __attribute__((ext_vector_type(2)));
typedef float v8f __attribute__((ext_vector_type(8)));

static __device__ __forceinline__ v8f wmma4(v2f a, v2f b, v8f c) {
  // D = A(16x4 f32) x B(4x16 f32) + C(16x16 f32)
  return __builtin_amdgcn_wmma_f32_16x16x4_f32(false, a, false, b, (short)0, c,
                                               false, false);
}

// Weights are stored K-paired: element (k, n) of a [K, 80] matrix lives at
// ((k>>1)*80 + n)*2 + (k&1), so a lane's (b.x, b.y) = (W[k0][f], W[k0+1][f])
// is one aligned 8-byte load into an even VGPR pair.
__device__ __forceinline__ int paired_idx(int k, int n) {
  return ((k >> 1) * FP + n) * 2 + (k & 1);
}

// ---------------- utility ----------------
__global__ void fill_kernel(float* __restrict__ p, float v, int n) {
  int i = blockIdx.x * blockDim.x + threadIdx.x;
  if (i < n) p[i] = v;
}

__device__ __forceinline__ void atomicMaxF(float* a, float v) {
  if (v >= 0.0f) atomicMax((int*)a, __float_as_int(v));
  else           atomicMin((unsigned int*)a, __float_as_uint(v));
}
__device__ __forceinline__ void atomicMinF(float* a, float v) {
  if (v >= 0.0f) atomicMin((int*)a, __float_as_int(v));
  else           atomicMax((unsigned int*)a, __float_as_uint(v));
}

// ---------------- weight preprocessing ----------------
// Pad pre_w [L,150,75] -> Wp [L, KE*FP] K-paired, pre_b -> bp [L,80]
__global__ void prep_pad_pre(const float* __restrict__ pre_w,
                             const float* __restrict__ pre_b,
                             float* __restrict__ Wp, float* __restrict__ bp) {
  int t = blockIdx.x * blockDim.x + threadIdx.x;
  int total = LL * KE * FP;
  if (t < total) {
    int l = t / (KE * FP);
    int r = t % (KE * FP);
    int k = r / FP, n = r % FP;
    float v = (k < 2 * FD && n < FD) ? pre_w[(l * 2 * FD + k) * FD + n] : 0.0f;
    Wp[l * KE * FP + paired_idx(k, n)] = v;
  }
  if (t < LL * FP) {
    int l = t / FP, n = t % FP;
    bp[t] = (n < FD) ? pre_b[l * FD + n] : 0.0f;
  }
}

// Combined node weight: Wc[l] = post_w[l] @ lin_w[l], K-paired [KN*FP]
__global__ void prep_combine(const float* __restrict__ post_w,
                             const float* __restrict__ lin_w,
                             float* __restrict__ Wc) {
  int t = blockIdx.x * blockDim.x + threadIdx.x;
  if (t >= LL * KN * FP) return;
  int l = t / (KN * FP);
  int r = t % (KN * FP);
  int k = r / FP, n = r % FP;
  float v = 0.0f;
  if (k < 13 * FD && n < FD) {
    const float* pw = post_w + (l * 13 * FD + k) * FD;
    const float* lw = lin_w + l * FD * FD + n;
    for (int i = 0; i < FD; ++i) v += pw[i] * lw[i * FD];
  }
  Wc[l * KN * FP + paired_idx(k, n)] = v;
}

// bc[l] = post_b[l] @ lin_w[l] + lin_b[l], padded to [80]
__global__ void prep_bias(const float* __restrict__ post_b,
                          const float* __restrict__ lin_w,
                          const float* __restrict__ lin_b,
                          float* __restrict__ bc) {
  int t = blockIdx.x * blockDim.x + threadIdx.x;
  if (t >= LL * FP) return;
  int l = t / FP, n = t % FP;
  float v = 0.0f;
  if (n < FD) {
    v = lin_b[l * FD + n];
    for (int i = 0; i < FD; ++i)
      v += post_b[l * FD + i] * lin_w[(l * FD + i) * FD + n];
  }
  bc[t] = v;
}

// ---------------- GCN stage ----------------
__global__ void count_deg(const int* __restrict__ col, float* __restrict__ cnt) {
  int e = blockIdx.x * blockDim.x + threadIdx.x;
  if (e < NE) atomicAdd(cnt + col[e], 1.0f);
}

__global__ void gcn_xw(const float* __restrict__ x, const float* __restrict__ w,
                       float* __restrict__ xw) {
  int t = blockIdx.x * blockDim.x + threadIdx.x;
  if (t >= NN * FP) return;
  int node = t / FP, f = t % FP;
  float v = 0.0f;
  if (f < FD) {
    for (int k = 0; k < 6; ++k) v += x[node * 6 + k] * w[k * FD + f];
  }
  xw[t] = v;
}

__global__ void gcn_scatter(const int* __restrict__ row, const int* __restrict__ col,
                            const float* __restrict__ cnt, const float* __restrict__ xw,
                            float* __restrict__ h) {
  int t = blockIdx.x * blockDim.x + threadIdx.x;
  if (t >= NE * FD) return;
  int e = t / FD, f = t % FD;
  int ri = row[e], ci = col[e];
  float nrm = rsqrtf(cnt[ri] + 1.0f) * rsqrtf(cnt[ci] + 1.0f);
  atomicAdd(h + ci * FP + f, nrm * xw[ri * FP + f]);
}

__global__ void gcn_self(const float* __restrict__ cnt, const float* __restrict__ xw,
                         const float* __restrict__ b, float* __restrict__ h) {
  int t = blockIdx.x * blockDim.x + threadIdx.x;
  if (t >= NN * FD) return;
  int i = t / FD, f = t % FD;
  h[i * FP + f] += xw[i * FP + f] / (cnt[i] + 1.0f) + b[f];
}

__global__ void avg_log_kernel(const float* __restrict__ cnt, float* __restrict__ avgbuf) {
  __shared__ float sm[256];
  int i = blockIdx.x * 256 + threadIdx.x;
  sm[threadIdx.x] = (i < NN) ? logf(cnt[i] + 1.0f) : 0.0f;
  __syncthreads();
  for (int s = 128; s > 0; s >>= 1) {
    if ((int)threadIdx.x < s) sm[threadIdx.x] += sm[threadIdx.x + s];
    __syncthreads();
  }
  if (threadIdx.x == 0) atomicAdd(avgbuf, sm[0]);
}

// ---------------- edge GEMM: m = [h[col], h[row]] @ Wp + bp ----------------
// grid: NE/128 blocks, 128 threads (4 waves), each wave = 32 edges (2 M-tiles).
__global__ __launch_bounds__(128)
void edge_gemm(const float* __restrict__ h, const int* __restrict__ row,
               const int* __restrict__ col, const float* __restrict__ Wp,
               const float* __restrict__ bp, float* __restrict__ m) {
  __shared__ float Bs[KE * FP];        // 48,640 B, K-paired weight copy
  __shared__ float As[4][32][KESTR];   // 79,872 B, per-wave A tiles

  for (int idx = threadIdx.x; idx < KE * FP; idx += 128) Bs[idx] = Wp[idx];

  const int wave = threadIdx.x >> 5;
  const int lane = threadIdx.x & 31;
  const int tile2 = blockIdx.x * 4 + wave;  // 32-edge tile

  // Stage A tile: 32 edges x [h[col](75) | h[row](75) | 0 0] (coalesced rows)
  for (int e2 = 0; e2 < 32; ++e2) {
    const int eg = tile2 * 32 + e2;
    const int ci = col[eg], ri = row[eg];
    const float* hc = h + ci * FP;
    const float* hr = h + ri * FP;
    for (int f = lane; f < KE; f += 32) {
      As[wave][e2][f] = (f < FD) ? hc[f] : ((f < 2 * FD) ? hr[f - FD] : 0.0f);
    }
  }
  __syncthreads();

  const int n = lane & 15;
  const int khalf = (lane >> 4) << 1;  // 0 or 2
  const int kp = lane >> 4;            // khalf >> 1

  v8f acc[10] = {};
  for (int kk = 0; kk < KE; kk += 4) {
    v2f a0 = *(const v2f*)&As[wave][n][kk + khalf];
    v2f a1 = *(const v2f*)&As[wave][16 + n][kk + khalf];
    const float* bbase = Bs + ((kk >> 1) + kp) * (FP * 2) + n * 2;
#pragma unroll
    for (int nt = 0; nt < 5; ++nt) {
      v2f b = *(const v2f*)(bbase + nt * 32);
      acc[nt] = wmma4(a0, b, acc[nt]);
      acc[5 + nt] = wmma4(a1, b, acc[5 + nt]);
    }
  }

  const int mbase = kp * 8;
  float* mp0 = m + (tile2 * 32 + mbase) * FP + n;
  float* mp1 = mp0 + 16 * FP;
#pragma unroll
  for (int nt = 0; nt < 5; ++nt) {
    const float bias = bp[nt * 16 + n];
#pragma unroll
    for (int r = 0; r < 8; ++r) {
      mp0[r * FP + nt * 16] = acc[nt][r] + bias;
      mp1[r * FP + nt * 16] = acc[5 + nt][r] + bias;
    }
  }
}

// ---------------- edge aggregation (sum/sumsq/min/max over col) ----------------
__global__ void edge_agg(const int* __restrict__ col, const float* __restrict__ m,
                         float* __restrict__ aS, float* __restrict__ aQ,
                         float* __restrict__ aMn, float* __restrict__ aMx) {
  int t = blockIdx.x * blockDim.x + threadIdx.x;
  if (t >= NE * FD) return;
  int e = t / FD, f = t % FD;
  float v = m[e * FP + f];
  int c = col[e];
  int idx = c * FP + f;
  atomicAdd(aS + idx, v);
  atomicAdd(aQ + idx, v * v);
  atomicMinF(aMn + idx, v);
  atomicMaxF(aMx + idx, v);
}

// ---------------- build hcat13 = [h, agg, agg*amp, agg*att] (padded 976) --------
__global__ void pna_finalize(const float* __restrict__ h, const float* __restrict__ cnt,
                             const float* __restrict__ avgbuf,
                             const float* __restrict__ aS, const float* __restrict__ aQ,
                             const float* __restrict__ aMn, const float* __restrict__ aMx,
                             float* __restrict__ hcat) {
  int t = blockIdx.x * blockDim.x + threadIdx.x;
  if (t >= NN * FD) return;
  int i = t / FD, f = t % FD;
  float c = cnt[i];
  float cc = fmaxf(c, 1.0f);
  int idx = i * FP + f;
  float mean = aS[idx] / cc;
  float msq = aQ[idx] / cc;
  float var = msq - mean * mean;
  if (var < 0.0f) var = 0.0f;
  float stdv = sqrtf(var + EPSV);
  bool has = c > 0.0f;
  float mn = has ? aMn[idx] : 0.0f;
  float mx = has ? aMx[idx] : 0.0f;
  float avgl = avgbuf[0] / (float)NN;
  float ld = logf(fmaxf(c, 1.0f) + 1.0f);
  float amp = ld / avgl;
  float att = avgl / ld;
  int base = i * KN;
  hcat[base + f] = h[i * FP + f];
  hcat[base + 1 * FD + f] = mean;
  hcat[base + 2 * FD + f] = mn;
  hcat[base + 3 * FD + f] = mx;
  hcat[base + 4 * FD + f] = stdv;
  hcat[base + 5 * FD + f] = mean * amp;
  hcat[base + 6 * FD + f] = mn * amp;
  hcat[base + 7 * FD + f] = mx * amp;
  hcat[base + 8 * FD + f] = stdv * amp;
  hcat[base + 9 * FD + f] = mean * att;
  hcat[base + 10 * FD + f] = mn * att;
  hcat[base + 11 * FD + f] = mx * att;
  hcat[base + 12 * FD + f] = stdv * att;
  if (f == 0) hcat[base + 13 * FD] = 0.0f;  // pad column 975
}

// ---------------- node GEMM: o = hcat13 @ Wc + bc ----------------
// each wave = 32 nodes (2 M-tiles); A direct from global, B (K-paired) via cache.
__global__ __launch_bounds__(128)
void node_gemm(const float* __restrict__ A, const float* __restrict__ W,
               const float* __restrict__ bias, float* __restrict__ o) {
  const int wave = threadIdx.x >> 5;
  const int lane = threadIdx.x & 31;
  const int tile2 = blockIdx.x * 4 + wave;
  if (tile2 >= NN / 32) return;  // wave-uniform
  const int n = lane & 15;
  const float* arow0 = A + (tile2 * 32 + n) * KN;
  const float* arow1 = arow0 + 16 * KN;
  const int khalf = (lane >> 4) << 1;
  const int kp = lane >> 4;

  v8f acc[10] = {};
  for (int kk = 0; kk < KN; kk += 4) {
    v2f a0 = *(const v2f*)(arow0 + kk + khalf);
    v2f a1 = *(const v2f*)(arow1 + kk + khalf);
    const float* bbase = W + ((kk >> 1) + kp) * (FP * 2) + n * 2;
#pragma unroll
    for (int nt = 0; nt < 5; ++nt) {
      v2f b = *(const v2f*)(bbase + nt * 32);
      acc[nt] = wmma4(a0, b, acc[nt]);
      acc[5 + nt] = wmma4(a1, b, acc[5 + nt]);
    }
  }

  const int mbase = kp * 8;
  float* op0 = o + (tile2 * 32 + mbase) * FP + n;
  float* op1 = op0 + 16 * FP;
#pragma unroll
  for (int nt = 0; nt < 5; ++nt) {
    const float bv = bias[nt * 16 + n];
#pragma unroll
    for (int r = 0; r < 8; ++r) {
      op0[r * FP + nt * 16] = acc[nt][r] + bv;
      op1[r * FP + nt * 16] = acc[5 + nt][r] + bv;
    }
  }
}

// ---------------- BatchNorm ----------------
__global__ void bn_stats(const float* __restrict__ o, float* __restrict__ bns) {
  int t = blockIdx.x * blockDim.x + threadIdx.x;
  if (t >= 256 * FP) return;
  int f = t % FP;
  int chunk = t / FP;
  if (f >= FD) return;
  float s = 0.0f, q = 0.0f;
  for (int i = chunk; i < NN; i += 256) {
    float v = o[i * FP + f];
    s += v;
    q += v * v;
  }
  atomicAdd(bns + f, s);
  atomicAdd(bns + FP + f, q);
}

__global__ void bn_apply(const float* __restrict__ o, const float* __restrict__ bns,
                         const float* __restrict__ g, const float* __restrict__ b,
                         int l, float* __restrict__ h) {
  int t = blockIdx.x * blockDim.x + threadIdx.x;
  if (t >= NN * FP) return;
  int i = t / FP, f = t % FP;
  float y = 0.0f;
  if (f < FD) {
    float mu = bns[f] / (float)NN;
    float var = bns[FP + f] / (float)NN - mu * mu;
    float v = o[i * FP + f];
    y = (v - mu) * rsqrtf(var + EPSV) * g[l * FD + f] + b[l * FD + f];
    y = fmaxf(y, 0.0f);
  }
  h[i * FP + f] = y;
}

// ---------------- pooling + MLP head ----------------
__global__ void pool_scatter(const int* __restrict__ batch, const float* __restrict__ h,
                             float* __restrict__ g) {
  int t = blockIdx.x * blockDim.x + threadIdx.x;
  if (t >= NN * FD) return;
  int i = t / FD, f = t % FD;
  atomicAdd(g + batch[i] * FP + f, h[i * FP + f]);
}

__global__ void mlp1(const float* __restrict__ g, const float* __restrict__ w,
                     const float* __restrict__ b, float* __restrict__ g1) {
  int t = blockIdx.x * blockDim.x + threadIdx.x;
  if (t >= GG * 50) return;
  int gi = t / 50, j = t % 50;
  float v = b[j];
  for (int k = 0; k < FD; ++k) v += g[gi * FP + k] * w[k * 50 + j];
  g1[gi * 64 + j] = fmaxf(v, 0.0f);
}

__global__ void mlp2(const float* __restrict__ g1, const float* __restrict__ w,
                     const float* __restrict__ b, float* __restrict__ g2) {
  int t = blockIdx.x * blockDim.x + threadIdx.x;
  if (t >= GG * 25) return;
  int gi = t / 25, j = t % 25;
  float v = b[j];
  for (int k = 0; k < 50; ++k) v += g1[gi * 64 + k] * w[k * 25 + j];
  g2[gi * 32 + j] = fmaxf(v, 0.0f);
}

__global__ void mlp3(const float* __restrict__ g2, const float* __restrict__ w,
                     const float* __restrict__ b, float* __restrict__ out) {
  int t = blockIdx.x * blockDim.x + threadIdx.x;
  if (t >= GG * CCL) return;
  int gi = t / CCL, j = t % CCL;
  float v = b[j];
  for (int k = 0; k < 25; ++k) v += g2[gi * 32 + k] * w[k * CCL + j];
  out[gi * CCL + j] = v;
}

// ---------------- host orchestration ----------------
extern "C" void kernel_launch(void* const* d_in, const int* in_sizes, int n_in,
                              void* d_out, int out_size, void* d_ws, size_t ws_size,
                              hipStream_t stream) {
  const float* x = (const float*)d_in[0];
  const int* eidx = (const int*)d_in[1];
  const int* batch = (const int*)d_in[2];
  const float* gcn_w = (const float*)d_in[3];
  const float* gcn_b = (const float*)d_in[4];
  const float* pre_w = (const float*)d_in[5];
  const float* pre_b = (const float*)d_in[6];
  const float* post_w = (const float*)d_in[7];
  const float* post_b = (const float*)d_in[8];
  const float* lin_w = (const float*)d_in[9];
  const float* lin_b = (const float*)d_in[10];
  const float* bn_g = (const float*)d_in[11];
  const float* bn_b = (const float*)d_in[12];
  const float* w1 = (const float*)d_in[13];
  const float* b1 = (const float*)d_in[14];
  const float* w2 = (const float*)d_in[15];
  const float* b2 = (const float*)d_in[16];
  const float* w3 = (const float*)d_in[17];
  const float* b3 = (const float*)d_in[18];
  const int* row = eidx;       // edge_index[0]
  const int* col = eidx + NE;  // edge_index[1]

  float* ws = (float*)d_ws;
  size_t off = 0;
  auto alloc = [&](size_t nf) {
    float* p = ws + off;
    off += (nf + 63) & ~(size_t)63;
    return p;
  };
  float* cnt = alloc(NN);
  float* avgbuf = alloc(8);
  float* xw = alloc((size_t)NN * FP);
  float* h = alloc((size_t)NN * FP);
  float* uni = alloc((size_t)NE * FP);  // m [E,80] aliases hcat13 [N,976]
  float* m = uni;
  float* hcat = uni;
  float* aS = alloc((size_t)NN * FP);
  float* aQ = alloc((size_t)NN * FP);
  float* aMn = alloc((size_t)NN * FP);
  float* aMx = alloc((size_t)NN * FP);
  float* o = alloc((size_t)NN * FP);
  float* bns = alloc(2 * FP);
  float* Wp = alloc((size_t)LL * KE * FP);
  float* bp = alloc(LL * FP);
  float* Wc = alloc((size_t)LL * KN * FP);
  float* bc = alloc(LL * FP);
  float* gp = alloc(GG * FP);
  float* g1 = alloc(GG * 64);
  float* g2 = alloc(GG * 32);

  const int T = 256;
  auto blocks = [](long n, int t) { return (unsigned)((n + t - 1) / t); };

  // weight preprocessing
  prep_pad_pre<<<blocks((long)LL * KE * FP, T), T, 0, stream>>>(pre_w, pre_b, Wp, bp);
  prep_combine<<<blocks((long)LL * KN * FP, T), T, 0, stream>>>(post_w, lin_w, Wc);
  prep_bias<<<blocks(LL * FP, T), T, 0, stream>>>(post_b, lin_w, lin_b, bc);

  // GCN stage
  fill_kernel<<<blocks(NN, T), T, 0, stream>>>(cnt, 0.0f, NN);
  fill_kernel<<<1, 8, 0, stream>>>(avgbuf, 0.0f, 8);
  count_deg<<<blocks(NE, T), T, 0, stream>>>(col, cnt);
  gcn_xw<<<blocks((long)NN * FP, T), T, 0, stream>>>(x, gcn_w, xw);
  fill_kernel<<<blocks((long)NN * FP, T), T, 0, stream>>>(h, 0.0f, NN * FP);
  gcn_scatter<<<blocks((long)NE * FD, T), T, 0, stream>>>(row, col, cnt, xw, h);
  gcn_self<<<blocks((long)NN * FD, T), T, 0, stream>>>(cnt, xw, gcn_b, h);
  avg_log_kernel<<<blocks(NN, 256), 256, 0, stream>>>(cnt, avgbuf);

  // 5 x PNA layer
  for (int l = 0; l < LL; ++l) {
    fill_kernel<<<blocks((long)NN * FP, T), T, 0, stream>>>(aS, 0.0f, NN * FP);
    fill_kernel<<<blocks((long)NN * FP, T), T, 0, stream>>>(aQ, 0.0f, NN * FP);
    fill_kernel<<<blocks((long)NN * FP, T), T, 0, stream>>>(aMn, 3.402823466e+38f, NN * FP);
    fill_kernel<<<blocks((long)NN * FP, T), T, 0, stream>>>(aMx, -3.402823466e+38f, NN * FP);
    fill_kernel<<<1, 2 * FP, 0, stream>>>(bns, 0.0f, 2 * FP);

    edge_gemm<<<NE / 128, 128, 0, stream>>>(h, row, col, Wp + (size_t)l * KE * FP,
                                            bp + l * FP, m);
    edge_agg<<<blocks((long)NE * FD, T), T, 0, stream>>>(col, m, aS, aQ, aMn, aMx);
    pna_finalize<<<blocks((long)NN * FD, T), T, 0, stream>>>(h, cnt, avgbuf, aS, aQ,
                                                             aMn, aMx, hcat);
    node_gemm<<<(NN / 32 + 3) / 4, 128, 0, stream>>>(hcat, Wc + (size_t)l * KN * FP,
                                                     bc + l * FP, o);
    bn_stats<<<blocks(256 * FP, T), T, 0, stream>>>(o, bns);
    bn_apply<<<blocks((long)NN * FP, T), T, 0, stream>>>(o, bns, bn_g, bn_b, l, h);
  }

  // pooling + head
  fill_kernel<<<blocks(GG * FP, T), T, 0, stream>>>(gp, 0.0f, GG * FP);
  pool_scatter<<<blocks((long)NN * FD, T), T, 0, stream>>>(batch, h, gp);
  mlp1<<<blocks(GG * 50, T), T, 0, stream>>>(gp, w1, b1, g1);
  mlp2<<<blocks(GG * 25, T), T, 0, stream>>>(g1, w2, b2, g2);
  mlp3<<<blocks(GG * CCL, T), T, 0, stream>>>(g2, w3, b3, (float*)d_out);
}